// LCA_79336635892392
// MI455X (gfx1250) — compile-verified
//
#include <hip/hip_runtime.h>

#define NEUR 4096
#define PIX  3072
#define BAT  32
#define NSTEPS 99               // reference runs NUM_STEPS - 1 iterations
#define ETA_F ((float)(0.001 / 0.03))

typedef __attribute__((ext_vector_type(2))) float v2f;
typedef __attribute__((ext_vector_type(8))) float v8f;

// D = A(16x4) * B(4x16) + C, f32 matrix pipe
__device__ __forceinline__ v8f wmma4(v2f a, v2f b, v8f c) {
    return __builtin_amdgcn_wmma_f32_16x16x4_f32(
        /*neg_a=*/false, a, /*neg_b=*/false, b,
        /*c_mod=*/(short)0, c, /*reuse_a=*/false, /*reuse_b=*/false);
}

// ---------------------------------------------------------------------------
// Zero u and a (iteration state), neuron-major [n*32 + m]
__global__ void __launch_bounds__(256)
lca_init_kernel(float* __restrict__ uT, float* __restrict__ aT0) {
    int i = blockIdx.x * 256 + threadIdx.x;   // 512 blocks -> 131072 threads
    uT[i]  = 0.0f;
    aT0[i] = 0.0f;
}

// ---------------------------------------------------------------------------
// bT[n*32+m] = sum_k x[m,k] * phi[k,n].  One wave per 16x16 tile (512 tiles).
__global__ void __launch_bounds__(256)
lca_b_kernel(const float* __restrict__ x, const float* __restrict__ phi,
             float* __restrict__ bT) {
    const int lane = threadIdx.x & 31;
    const int wv   = threadIdx.x >> 5;
    const int tile = blockIdx.x * 8 + wv;       // 0..511
    const int m0   = (tile >> 8) * 16;          // 2 M tiles
    const int n0   = (tile & 255) * 16;         // 256 N tiles
    const int hk   = lane >> 4;                 // half-wave selects K pair
    const int l16  = lane & 15;

    const float* pA = x + (size_t)(m0 + l16) * PIX;   // row m of x
    const float* pB = phi + n0 + l16;                  // col n of phi
    v8f acc = {};
    for (int k = 0; k < PIX; k += 4) {
        const int ka = k + hk * 2;
        v2f a, b;
        a.x = pA[ka];
        a.y = pA[ka + 1];
        b.x = pB[(size_t)ka * NEUR];
        b.y = pB[(size_t)(ka + 1) * NEUR];
        acc = wmma4(a, b, acc);
    }
#pragma unroll
    for (int v = 0; v < 8; ++v) {
        const int m = m0 + v + hk * 8;
        const int n = n0 + l16;
        bT[n * BAT + m] = acc[v];
    }
}

// ---------------------------------------------------------------------------
// g[i,j] = sum_k phi[k,i]*phi[k,j] - (i==j).
// One wave per 32(i) x 64(j) tile: 2 A frags x 4 B frags -> 8 WMMA / k-step.
__global__ void __launch_bounds__(256)
lca_g_kernel(const float* __restrict__ phi, float* __restrict__ g) {
    const int lane  = threadIdx.x & 31;
    const int wv    = threadIdx.x >> 5;
    const int strip = blockIdx.x * 8 + wv;      // 0..8191
    const int i0 = (strip >> 6) * 32;           // 128 i blocks of 32 rows
    const int j0 = (strip & 63) * 64;           // 64 strips of width 64
    const int hk  = lane >> 4;
    const int l16 = lane & 15;

    const float* pA = phi + i0 + l16;           // A = phi^T: row-contiguous loads
    const float* pB = phi + j0 + l16;
    v8f c00 = {}, c01 = {}, c02 = {}, c03 = {};
    v8f c10 = {}, c11 = {}, c12 = {}, c13 = {};
    for (int k = 0; k < PIX; k += 4) {
        const int ka = k + hk * 2;
        const float* pa = pA + (size_t)ka * NEUR;
        const float* pb = pB + (size_t)ka * NEUR;
        v2f a0, a1, b0, b1, b2, b3;
        a0.x = pa[0];        a0.y = pa[NEUR];
        a1.x = pa[16];       a1.y = pa[NEUR + 16];
        b0.x = pb[0];        b0.y = pb[NEUR];
        b1.x = pb[16];       b1.y = pb[NEUR + 16];
        b2.x = pb[32];       b2.y = pb[NEUR + 32];
        b3.x = pb[48];       b3.y = pb[NEUR + 48];
        c00 = wmma4(a0, b0, c00);
        c01 = wmma4(a0, b1, c01);
        c02 = wmma4(a0, b2, c02);
        c03 = wmma4(a0, b3, c03);
        c10 = wmma4(a1, b0, c10);
        c11 = wmma4(a1, b1, c11);
        c12 = wmma4(a1, b2, c12);
        c13 = wmma4(a1, b3, c13);
    }
#pragma unroll
    for (int t = 0; t < 2; ++t) {
        v8f r0 = t ? c10 : c00, r1 = t ? c11 : c01;
        v8f r2 = t ? c12 : c02, r3 = t ? c13 : c03;
#pragma unroll
        for (int v = 0; v < 8; ++v) {
            const int i = i0 + t * 16 + v + hk * 8;
            const int j = j0 + l16;
            float e0 = r0[v], e1 = r1[v], e2 = r2[v], e3 = r3[v];
            if (i == j)      e0 -= 1.0f;
            if (i == j + 16) e1 -= 1.0f;
            if (i == j + 32) e2 -= 1.0f;
            if (i == j + 48) e3 -= 1.0f;
            float* go = g + (size_t)i * NEUR + j;
            go[0]  = e0;
            go[16] = e1;
            go[32] = e2;
            go[48] = e3;
        }
    }
}

// ---------------------------------------------------------------------------
// One LCA step. Block = one 16-col tile of the 32x4096 state, full batch M=32.
// 8 waves split K; each wave computes BOTH M-tiles sharing one B fragment,
// so g is streamed from L2 exactly once per step.
__global__ void __launch_bounds__(256)
lca_step_kernel(const float* __restrict__ aT_in, float* __restrict__ aT_out,
                float* __restrict__ uT, const float* __restrict__ bT,
                const float* __restrict__ g, const float* __restrict__ lam_p) {
    __shared__ float red[8 * 512];              // 16 KB partials
    const int lane = threadIdx.x & 31;
    const int wv   = threadIdx.x >> 5;          // 8 waves = 8 K slices
    const int n0   = blockIdx.x * 16;           // 256 blocks
    const int hk   = lane >> 4;
    const int l16  = lane & 15;

    v8f acc0 = {}, acc1 = {};
    const int kbeg = wv * (NEUR / 8);           // 512 K per wave
    const float* pA = aT_in + l16;              // neuron-major a: contiguous
    const float* pB = g + n0 + l16;
    for (int k = kbeg; k < kbeg + NEUR / 8; k += 4) {
        const int ka = k + hk * 2;
        const float* pa = pA + ka * BAT;
        v2f a0, a1, b;
        a0.x = pa[0];
        a0.y = pa[BAT];
        a1.x = pa[16];
        a1.y = pa[BAT + 16];
        const float* pb = pB + (size_t)ka * NEUR;
        __builtin_prefetch(pb + (size_t)16 * NEUR, 0, 1);
        b.x = pb[0];
        b.y = pb[NEUR];
        acc0 = wmma4(a0, b, acc0);
        acc1 = wmma4(a1, b, acc1);
    }
#pragma unroll
    for (int v = 0; v < 8; ++v) {
        const int e = (v + hk * 8) * 16 + l16;  // element index in 16x16 tile
        red[wv * 512 + e]       = acc0[v];
        red[wv * 512 + 256 + e] = acc1[v];
    }
    __syncthreads();

    const float lam = *lam_p;
    for (int e = threadIdx.x; e < 512; e += 256) {
        float s = red[e];
#pragma unroll
        for (int w = 1; w < 8; ++w) s += red[w * 512 + e];
        const int m = e >> 4;                   // 0..31 across both tiles
        const int n = n0 + (e & 15);
        const int idx = n * BAT + m;
        const float u  = uT[idx];
        const float un = u + ETA_F * (bT[idx] - s - u);
        uT[idx] = un;
        const float an = (un > lam) ? (un - lam)
                       : (un < -lam) ? (un + lam) : 0.0f;
        aT_out[idx] = an;
    }
}

// ---------------------------------------------------------------------------
// Transpose neuron-major a back to [batch, neurons] output.
__global__ void __launch_bounds__(256)
lca_out_kernel(const float* __restrict__ aT, float* __restrict__ out) {
    int t = blockIdx.x * 256 + threadIdx.x;     // 131072 threads
    int n = t & (NEUR - 1);
    int m = t >> 12;
    out[(size_t)m * NEUR + n] = aT[n * BAT + m];
}

// ---------------------------------------------------------------------------
extern "C" void kernel_launch(void* const* d_in, const int* in_sizes, int n_in,
                              void* d_out, int out_size, void* d_ws, size_t ws_size,
                              hipStream_t stream) {
    const float* x   = (const float*)d_in[0];   // [32, 3072]
    const float* phi = (const float*)d_in[1];   // [3072, 4096]
    const float* lam = (const float*)d_in[2];   // scalar

    float* g   = (float*)d_ws;                  // 4096*4096
    float* bT  = g   + (size_t)NEUR * NEUR;     // 4096*32 each below
    float* uT  = bT  + NEUR * BAT;
    float* aT0 = uT  + NEUR * BAT;
    float* aT1 = aT0 + NEUR * BAT;
    float* out = (float*)d_out;

    hipLaunchKernelGGL(lca_init_kernel, dim3(512),  dim3(256), 0, stream, uT, aT0);
    hipLaunchKernelGGL(lca_b_kernel,    dim3(64),   dim3(256), 0, stream, x, phi, bT);
    hipLaunchKernelGGL(lca_g_kernel,    dim3(1024), dim3(256), 0, stream, phi, g);

    float* bufs[2] = {aT0, aT1};
    for (int s = 0; s < NSTEPS; ++s) {
        hipLaunchKernelGGL(lca_step_kernel, dim3(256), dim3(256), 0, stream,
                           bufs[s & 1], bufs[(s + 1) & 1], uT, bT, g, lam);
    }
    hipLaunchKernelGGL(lca_out_kernel, dim3(512), dim3(256), 0, stream,
                       bufs[NSTEPS & 1], out);
}